// Sem_scal_loss_7610682049190
// MI455X (gfx1250) — compile-verified
//
#include <hip/hip_runtime.h>
#include <hip/hip_bf16.h>

typedef __attribute__((ext_vector_type(2))) float v2f;
typedef __attribute__((ext_vector_type(8))) float v8f;

#define NCLS 20
#define NCH 64
#define LOG_CLAMP 100.0f

// ---------------------------------------------------------------------------
// Kernel 1: streaming softmax + per-block partial reductions.
// Each thread owns 4 consecutive voxels -> global_load_b128 per class plane.
// Writes partial[block][64] = { sum_p[20], nominator[20], ct_count[20],
//                               n_masked, 0, 0, 0 }.
// ---------------------------------------------------------------------------
__global__ __launch_bounds__(256) void sem_partial_kernel(
    const float* __restrict__ pred, const int* __restrict__ target,
    float* __restrict__ partial, int N) {
  __shared__ float lds[NCH];
  const int tid = threadIdx.x;
  if (tid < NCH) lds[tid] = 0.0f;
  __syncthreads();

  float sp[NCLS];
#pragma unroll
  for (int c = 0; c < NCLS; ++c) sp[c] = 0.0f;
  float nmask = 0.0f;

  const int gtid = blockIdx.x * blockDim.x + tid;
  const int gstride = gridDim.x * blockDim.x;
  // Vector path only when the class-plane stride keeps float4 alignment.
  const int Ng = ((N & 3) == 0) ? (N >> 2) : 0;

  const float4* __restrict__ pred4 = (const float4*)pred;
  const int4* __restrict__ targ4 = (const int4*)target;

  for (int g = gtid; g < Ng; g += gstride) {
    const int4 t4 = targ4[g];
    const int tt[4] = {t4.x, t4.y, t4.z, t4.w};
    float ec[NCLS][4];
#pragma unroll
    for (int c = 0; c < NCLS; ++c) {
      const float4 r = pred4[(size_t)c * (size_t)Ng + (size_t)g];
      ec[c][0] = r.x; ec[c][1] = r.y; ec[c][2] = r.z; ec[c][3] = r.w;
    }
#pragma unroll
    for (int j = 0; j < 4; ++j) {
      const int t = tt[j];
      if (t >= 0 && t < NCLS) {  // IGNORE_INDEX (255) fails this
        float mx = ec[0][j];
#pragma unroll
        for (int c = 1; c < NCLS; ++c) mx = fmaxf(mx, ec[c][j]);
        float s = 0.0f, pt = 0.0f;
#pragma unroll
        for (int c = 0; c < NCLS; ++c) {
          const float x = __expf(ec[c][j] - mx);
          ec[c][j] = x;
          s += x;
          pt = (c == t) ? x : pt;  // p at target class, no dynamic indexing
        }
        const float inv = 1.0f / s;
#pragma unroll
        for (int c = 0; c < NCLS; ++c) sp[c] += ec[c][j] * inv;
        atomicAdd(&lds[NCLS + t], pt * inv);   // nominator[t] += p_t
        atomicAdd(&lds[2 * NCLS + t], 1.0f);   // ct_count[t]  += 1
        nmask += 1.0f;
      }
    }
  }

  // Scalar tail (covers everything when N % 4 != 0).
  for (int v = Ng * 4 + gtid; v < N; v += gstride) {
    const int t = target[v];
    if (t >= 0 && t < NCLS) {
      float e[NCLS];
      float mx = -3.402823466e+38f;
#pragma unroll
      for (int c = 0; c < NCLS; ++c) {
        e[c] = pred[(size_t)c * (size_t)N + (size_t)v];
        mx = fmaxf(mx, e[c]);
      }
      float s = 0.0f, pt = 0.0f;
#pragma unroll
      for (int c = 0; c < NCLS; ++c) {
        const float x = __expf(e[c] - mx);
        e[c] = x;
        s += x;
        pt = (c == t) ? x : pt;
      }
      const float inv = 1.0f / s;
#pragma unroll
      for (int c = 0; c < NCLS; ++c) sp[c] += e[c] * inv;
      atomicAdd(&lds[NCLS + t], pt * inv);
      atomicAdd(&lds[2 * NCLS + t], 1.0f);
      nmask += 1.0f;
    }
  }

  // wave32 cross-lane reduction of the dense accumulators
#pragma unroll
  for (int off = 16; off > 0; off >>= 1) {
#pragma unroll
    for (int c = 0; c < NCLS; ++c) sp[c] += __shfl_xor(sp[c], off, 32);
    nmask += __shfl_xor(nmask, off, 32);
  }
  if ((tid & 31) == 0) {
#pragma unroll
    for (int c = 0; c < NCLS; ++c) atomicAdd(&lds[c], sp[c]);
    atomicAdd(&lds[60], nmask);
  }
  __syncthreads();
  if (tid < NCH) partial[(size_t)blockIdx.x * NCH + tid] = lds[tid];
}

// ---------------------------------------------------------------------------
// Kernel 2: one wave. Reduce partial[NB][64] -> totals[64] with
// V_WMMA_F32_16X16X4_F32 (B = ones => exact f32 row sums), then compute the
// final clamped-log loss scalar.
//
// A-matrix f32 16x4 layout (ISA 7.12.2): lane L holds A[L&15][2*(L>>4)] in
// element 0 and A[L&15][2*(L>>4)+1] in element 1. With A[row][k] =
// partial[(cb+k)*64 + chan_base + row], each WMMA folds 4 block-records
// into the 16x16 accumulator (every column of D equals the channel total).
// ---------------------------------------------------------------------------
__global__ __launch_bounds__(32) void sem_finalize_kernel(
    const float* __restrict__ partial, float* __restrict__ out, int NB) {
  __shared__ float totals[NCH];
  __shared__ float acc[2];
  const int lane = threadIdx.x;

  v8f c0 = {}, c1 = {}, c2 = {}, c3 = {};
  v2f ones;
  ones[0] = 1.0f;
  ones[1] = 1.0f;

  const int row = lane & 15;
  const int kb = (lane >> 4) << 1;  // 0 for lanes 0-15, 2 for lanes 16-31

  for (int cb = 0; cb < NB; cb += 4) {
    const size_t base = (size_t)(cb + kb) * NCH;
    v2f a0, a1, a2, a3;
    a0[0] = partial[base + row];
    a0[1] = partial[base + NCH + row];
    a1[0] = partial[base + 16 + row];
    a1[1] = partial[base + NCH + 16 + row];
    a2[0] = partial[base + 32 + row];
    a2[1] = partial[base + NCH + 32 + row];
    a3[0] = partial[base + 48 + row];
    a3[1] = partial[base + NCH + 48 + row];
    c0 = __builtin_amdgcn_wmma_f32_16x16x4_f32(false, a0, false, ones, (short)0, c0, false, false);
    c1 = __builtin_amdgcn_wmma_f32_16x16x4_f32(false, a1, false, ones, (short)0, c1, false, false);
    c2 = __builtin_amdgcn_wmma_f32_16x16x4_f32(false, a2, false, ones, (short)0, c2, false, false);
    c3 = __builtin_amdgcn_wmma_f32_16x16x4_f32(false, a3, false, ones, (short)0, c3, false, false);
  }

  // D layout: VGPR r, lanes 0-15 -> M=r ; lanes 16-31 -> M=8+r. Column value
  // is identical for every N, so lane 0 / lane 16 each own 8 rows per chain.
  if (lane == 0) {
#pragma unroll
    for (int r = 0; r < 8; ++r) {
      totals[r] = c0[r];
      totals[16 + r] = c1[r];
      totals[32 + r] = c2[r];
      totals[48 + r] = c3[r];
    }
  }
  if (lane == 16) {
#pragma unroll
    for (int r = 0; r < 8; ++r) {
      totals[8 + r] = c0[r];
      totals[24 + r] = c1[r];
      totals[40 + r] = c2[r];
      totals[56 + r] = c3[r];
    }
  }
  if (lane < 2) acc[lane] = 0.0f;
  __syncthreads();

  if (lane < NCLS) {
    const float sum_p = totals[lane];
    const float nom = totals[NCLS + lane];
    const float ct = totals[2 * NCLS + lane];
    const float nmask = totals[60];
    const bool valid = ct > 0.0f;
    const float neg_count = nmask - ct;
    const float spec_num = nmask - sum_p - ct + nom;
    const float prec = nom / fmaxf(sum_p, 1e-38f);
    const float rec = nom / fmaxf(ct, 1.0f);
    const float spec = spec_num / fmaxf(neg_count, 1.0f);
    float l = 0.0f;
    if (valid && (sum_p > 0.0f))     l += fminf(-logf(fmaxf(prec, 1e-38f)), LOG_CLAMP);
    if (valid)                        l += fminf(-logf(fmaxf(rec, 1e-38f)), LOG_CLAMP);
    if (valid && (neg_count > 0.0f)) l += fminf(-logf(fmaxf(spec, 1e-38f)), LOG_CLAMP);
    atomicAdd(&acc[0], l);
    atomicAdd(&acc[1], valid ? 1.0f : 0.0f);
  }
  __syncthreads();
  if (lane == 0) out[0] = acc[0] / acc[1];
}

// ---------------------------------------------------------------------------
extern "C" void kernel_launch(void* const* d_in, const int* in_sizes, int n_in,
                              void* d_out, int out_size, void* d_ws, size_t ws_size,
                              hipStream_t stream) {
  const float* pred = (const float*)d_in[0];
  const int* target = (const int*)d_in[1];
  float* out = (float*)d_out;
  float* partial = (float*)d_ws;

  const int N = in_sizes[1];  // number of voxels (B*H*W*D)

  int NB = 1024;  // 1024 blocks x 8 waves: plenty of HBM-level parallelism
  const size_t need = (size_t)NB * NCH * sizeof(float);
  if (ws_size < need) {
    NB = (int)(ws_size / (NCH * sizeof(float)));
    NB &= ~3;            // kernel 2 consumes records 4 at a time
    if (NB < 4) NB = 4;
  }

  sem_partial_kernel<<<NB, 256, 0, stream>>>(pred, target, partial, N);
  sem_finalize_kernel<<<1, 32, 0, stream>>>(partial, out, NB);
}